// XformModule_13615046328613
// MI455X (gfx1250) — compile-verified
//
#include <hip/hip_runtime.h>
#include <hip/hip_bf16.h>

// Problem constants (match reference setup_inputs)
constexpr int B = 4, I = 512, K = 512, C = 64;
constexpr float EPS_LN = 1e-5f;

typedef float v2f __attribute__((ext_vector_type(2)));
typedef float v8f __attribute__((ext_vector_type(8)));

// ---------------------------------------------------------------------------
// Kernel: D[M,64] = A[M,64] @ W[64,64]^T + bias (+ optional residual)
// Exact f32 GEMM on the CDNA5 matrix core: V_WMMA_F32_16X16X4_F32.
// One wave computes one 16x16 output tile; 16 k-steps of depth 4 over C=64.
// A 16x4 operand layout (ISA 7.12.2): lanes 0-15 -> K=0,1 in v[0],v[1];
// lanes 16-31 -> K=2,3.  B 4x16 mirrors this with N striped across lanes.
// C/D layout: VGPR v, lanes 0-15 -> M=v; lanes 16-31 -> M=v+8.
// ---------------------------------------------------------------------------
__global__ __launch_bounds__(256) void gemm_nt64_wmma(
    const float* __restrict__ A,      // [M,64]
    const float* __restrict__ W,      // [64,64] (we compute A @ W^T)
    const float* __restrict__ bias,   // [64]
    const float* __restrict__ resid,  // [M,64] or nullptr
    float* __restrict__ D,            // [M,64]
    int M)
{
    const int wave = threadIdx.x >> 5;
    const int lane = threadIdx.x & 31;
    const int tile = blockIdx.x * 8 + wave;       // 8 waves per block
    const int tm = tile >> 2;                     // 4 N-tiles (64/16)
    const int tn = tile & 3;
    if (tm * 16 >= M) return;                     // wave-uniform -> EXEC stays all-1s
    const int half = lane >> 4;                   // 0: lanes 0-15, 1: lanes 16-31
    const int r    = lane & 15;
    const int m0 = tm * 16, n0 = tn * 16;

    const float* arow = A + (size_t)(m0 + r) * 64;   // A[m0+r, :]
    const float* brow = W + (size_t)(n0 + r) * 64;   // W[n0+r, :]  (B[k][n] = W[n][k])

    v8f acc = {};
    for (int kk = 0; kk < 64; kk += 4) {
        v2f a, b;
        a.x = arow[kk + 2 * half + 0];
        a.y = arow[kk + 2 * half + 1];
        b.x = brow[kk + 2 * half + 0];
        b.y = brow[kk + 2 * half + 1];
        acc = __builtin_amdgcn_wmma_f32_16x16x4_f32(
            /*neg_a=*/false, a, /*neg_b=*/false, b,
            /*c_mod=*/(short)0, acc, /*reuse_a=*/false, /*reuse_b=*/false);
    }

    const int col = n0 + r;
    const float bv = bias[col];
    #pragma unroll
    for (int v = 0; v < 8; ++v) {
        const int row = m0 + v + 8 * half;
        float val = acc[v] + bv;
        if (resid) val += resid[(size_t)row * 64 + col];
        D[(size_t)row * 64 + col] = val;
    }
}

// ---------------------------------------------------------------------------
// Kernel: S[b,k,c] = sum_i exp(z[b,i,k,c]) * (mask[b,i,k] + 1e-6),
// then in-place G = Qa / S.   One thread per (b,k,c); coalesced over k*C+c,
// strided loop over i (stride K*C).  First full pass over z (268 MB).
// ---------------------------------------------------------------------------
__global__ __launch_bounds__(256) void colsum_div_kernel(
    const float* __restrict__ z,     // [B,I,K,C]
    const float* __restrict__ mask,  // [B,I,K]
    float* __restrict__ QaG)         // in: Qa [B,K,C]; out: G = Qa/S
{
    const int idx = blockIdx.x * 256 + threadIdx.x;   // over B*K*C
    const int c = idx & (C - 1);
    const int k = (idx / C) & (K - 1);
    const int b = idx / (C * K);
    const float* zp = z    + (size_t)b * I * K * C + (size_t)k * C + c;
    const float* mp = mask + (size_t)b * I * K + k;
    float s = 0.f;
    for (int i = 0; i < I; ++i) {
        const float w = mp[(size_t)i * K];
        s += __expf(zp[(size_t)i * (K * C)]) * (w + 1e-6f);
    }
    QaG[idx] = QaG[idx] / s;
}

// ---------------------------------------------------------------------------
// Kernel: Va[b,i,c] = sum_k mask*(mask+1e-6)*exp(z[b,i,k,c]) * G[b,k,c]
// One block per (b,i); mask row staged in LDS; threads split as
// (kq = tid/64 in 0..3) x (c = tid%64); LDS reduction over kq.
// Second full pass over z (268 MB); G per batch (128 KB) lives in L2.
// ---------------------------------------------------------------------------
__global__ __launch_bounds__(256) void va_kernel(
    const float* __restrict__ z,     // [B,I,K,C]
    const float* __restrict__ mask,  // [B,I,K]
    const float* __restrict__ G,     // [B,K,C]
    float* __restrict__ Va)          // [B,I,C]
{
    __shared__ float msk[K];
    __shared__ float red[256];
    const int bi  = blockIdx.x;          // b*I + i
    const int b   = bi / I;
    const int tid = threadIdx.x;

    msk[tid]       = mask[(size_t)bi * K + tid];
    msk[tid + 256] = mask[(size_t)bi * K + tid + 256];
    __syncthreads();

    const int c  = tid & 63;
    const int kq = tid >> 6;             // 0..3
    const float* zp = z + (size_t)bi * K * C + c;
    const float* gp = G + (size_t)b  * K * C + c;

    float acc = 0.f;
    for (int k = kq; k < K; k += 4) {
        const float w = msk[k];
        const float f = w * (w + 1e-6f); // mask * (mask + 1e-6)
        acc += f * __expf(zp[(size_t)k * C]) * gp[(size_t)k * C];
    }
    red[tid] = acc;
    __syncthreads();
    if (kq == 0) {
        Va[(size_t)bi * C + c] = red[c] + red[c + 64] + red[c + 128] + red[c + 192];
    }
}

// ---------------------------------------------------------------------------
// Kernel: layer-norm over the last two dims (I*C = 32768 elems per batch).
// One block per batch; block reduction of sum/sumsq; safe in-place.
// ---------------------------------------------------------------------------
__global__ __launch_bounds__(256) void ln2_kernel(
    const float* __restrict__ in, float* __restrict__ out)
{
    __shared__ float s1[256], s2[256];
    constexpr int N = I * C;
    const int b = blockIdx.x;
    const float* p = in + (size_t)b * N;
    float s = 0.f, ss = 0.f;
    for (int idx = threadIdx.x; idx < N; idx += 256) {
        const float x = p[idx];
        s += x; ss += x * x;
    }
    s1[threadIdx.x] = s; s2[threadIdx.x] = ss;
    __syncthreads();
    for (int off = 128; off > 0; off >>= 1) {
        if (threadIdx.x < off) {
            s1[threadIdx.x] += s1[threadIdx.x + off];
            s2[threadIdx.x] += s2[threadIdx.x + off];
        }
        __syncthreads();
    }
    const float mu   = s1[0] * (1.f / N);
    const float var  = s2[0] * (1.f / N) - mu * mu;
    const float rstd = rsqrtf(var + EPS_LN);
    float* o = out + (size_t)b * N;
    for (int idx = threadIdx.x; idx < N; idx += 256) {
        o[idx] = (p[idx] - mu) * rstd;
    }
}

// ---------------------------------------------------------------------------
// Launch pipeline (dim assumed == 1 per setup_inputs)
// ---------------------------------------------------------------------------
extern "C" void kernel_launch(void* const* d_in, const int* in_sizes, int n_in,
                              void* d_out, int out_size, void* d_ws, size_t ws_size,
                              hipStream_t stream) {
    (void)in_sizes; (void)n_in; (void)out_size; (void)ws_size;
    const float* V  = (const float*)d_in[0];
    const float* Q  = (const float*)d_in[1];
    const float* z  = (const float*)d_in[2];
    const float* zm = (const float*)d_in[3];
    const float* W1 = (const float*)d_in[4];
    const float* b1 = (const float*)d_in[5];
    const float* W2 = (const float*)d_in[6];
    const float* b2 = (const float*)d_in[7];
    float* out = (float*)d_out;

    float* Qa = (float*)d_ws;          // [B,K,C] -> becomes G after kernel 2
    float* Va = Qa + (size_t)B * K * C; // [B,I,C]
    float* Vt = Va + (size_t)B * I * C; // [B,I,C]

    // 1) Qa = Q @ W1^T + b1      (WMMA f32, M = B*K = 2048 -> 512 tiles / 8 waves)
    gemm_nt64_wmma<<<64, 256, 0, stream>>>(Q, W1, b1, nullptr, Qa, B * K);

    // 2) G = Qa / S  (first pass over z)
    colsum_div_kernel<<<(B * K * C) / 256, 256, 0, stream>>>(z, zm, Qa);

    // 3) Va (second pass over z)
    va_kernel<<<B * I, 256, 0, stream>>>(z, zm, Qa, Va);

    // 4) LN over (I,C), in place
    ln2_kernel<<<B, 256, 0, stream>>>(Va, Va);

    // 5) Vt = V + Va @ W2^T + b2 (WMMA f32, residual fused)
    gemm_nt64_wmma<<<64, 256, 0, stream>>>(Va, W2, b2, V, Vt, B * I);

    // 6) final LN -> out
    ln2_kernel<<<B, 256, 0, stream>>>(Vt, out);
}